// SpatialMQA_11424613008053
// MI455X (gfx1250) — compile-verified
//
#include <hip/hip_runtime.h>
#include <math.h>

typedef __attribute__((ext_vector_type(2))) float v2f;
typedef __attribute__((ext_vector_type(8))) float v8f;

#define BATCH   16
#define EMB     1024
#define HD      256
#define KV_COLS 512
#define V_OFF   256     // v = columns [256, 512) of W_kv
#define HW      4096    // 64*64 spatial positions per batch

// ---------------------------------------------------------------------------
// Kernel 1: one workgroup, 16 wave32s.
// Wave w computes the 16x16 tile v[0:16, w*16:(w+1)*16] = condition @ W_kv_v
// via V_WMMA_F32_16X16X4_F32 accumulated over K=1024 (256 WMMAs), then fuses
// the per-row dot with W_score, reduces across lanes/waves, applies bias +
// sigmoid, and writes 16 per-batch scores to workspace.
// ---------------------------------------------------------------------------
__global__ __launch_bounds__(512) void mqa_scores_kernel(
    const float* __restrict__ cond,     // (16, 1024)
    const float* __restrict__ W_kv,     // (1024, 512) row-major
    const float* __restrict__ W_score,  // (256, 1)
    const float* __restrict__ b_score,  // (1,)
    float* __restrict__ sig_out)        // (16,) sigmoid(score) per batch
{
    __shared__ float lds_cond[BATCH * EMB];   // 64 KB staging of condition
    __shared__ float lds_part[16][16];        // [wave][batch-row] partials

    const int tid  = threadIdx.x;
    const int wave = tid >> 5;
    const int lane = tid & 31;

    // Stage condition into LDS once (512 threads x 32 floats each).
    for (int i = tid; i < BATCH * EMB; i += 512)
        lds_cond[i] = cond[i];
    __syncthreads();

    const int half = lane >> 4;     // 0 = lanes 0-15, 1 = lanes 16-31
    const int l16  = lane & 15;
    const int koff = half * 2;      // K offset carried by upper lane half

    // W_kv column this lane feeds for the B matrix (v-half of W_kv).
    const int ncol = V_OFF + wave * 16 + l16;

    v8f acc = {};
    for (int k = 0; k < EMB; k += 4) {
        // A: 16x4 f32 tile of condition. lanes hold M=l16; VGPR0/1 = K pairs.
        v2f a, b;
        a.x = lds_cond[l16 * EMB + k + koff];
        a.y = lds_cond[l16 * EMB + k + koff + 1];
        // B: 4x16 f32 tile of W_kv. VGPR0 = rows K0/K2, VGPR1 = rows K1/K3.
        b.x = W_kv[(k + koff)     * KV_COLS + ncol];
        b.y = W_kv[(k + koff + 1) * KV_COLS + ncol];
        // D = A*B + C  (8 args: neg_a, A, neg_b, B, c_mod, C, reuse_a, reuse_b)
        acc = __builtin_amdgcn_wmma_f32_16x16x4_f32(
            false, a, false, b, (short)0, acc, false, false);
    }

    // Fused score: this wave's 16 head-dim columns dotted with W_score.
    // acc[i] = v[M, N] with M = i + half*8, N = l16 (within wave's 16 cols).
    const float ws = W_score[wave * 16 + l16];
    float p[8];
    #pragma unroll
    for (int i = 0; i < 8; ++i) p[i] = acc[i] * ws;

    // Butterfly-reduce over the 16 lanes of each half (wave32 permutes).
    #pragma unroll
    for (int m = 1; m <= 8; m <<= 1) {
        #pragma unroll
        for (int i = 0; i < 8; ++i)
            p[i] += __shfl_xor(p[i], m, 32);
    }

    // Lane 0 of each half owns rows [half*8, half*8+8).
    if (l16 == 0) {
        #pragma unroll
        for (int i = 0; i < 8; ++i)
            lds_part[wave][half * 8 + i] = p[i];
    }
    __syncthreads();

    // Cross-wave reduce: 16 threads, one per batch row.
    if (tid < BATCH) {
        float s = b_score[0];
        #pragma unroll
        for (int w = 0; w < 16; ++w) s += lds_part[w][tid];
        sig_out[tid] = 1.0f / (1.0f + __expf(-s));
    }
}

// ---------------------------------------------------------------------------
// Kernel 2: broadcast the 16 per-batch scalars over (B, 1, 64, 64).
// float4 stores: 65536 floats = 16384 vec4s.
// ---------------------------------------------------------------------------
__global__ __launch_bounds__(256) void mqa_broadcast_kernel(
    const float* __restrict__ sig,  // (16,)
    float4* __restrict__ out)       // (B*HW/4,) vec4 view
{
    const int idx = blockIdx.x * 256 + threadIdx.x;   // 0 .. 16383
    const int b   = idx >> 10;                        // 1024 vec4s per batch
    const float s = sig[b];
    out[idx] = make_float4(s, s, s, s);
}

extern "C" void kernel_launch(void* const* d_in, const int* in_sizes, int n_in,
                              void* d_out, int out_size, void* d_ws, size_t ws_size,
                              hipStream_t stream) {
    // Inputs (setup_inputs order): x, condition, W_q, W_kv, W_score, b_score.
    // x and W_q are mathematically dead (softmax over a length-1 axis == 1),
    // so we never touch d_in[0]/d_in[2]: saves 64 MB of HBM traffic.
    const float* cond    = (const float*)d_in[1];
    const float* W_kv    = (const float*)d_in[3];
    const float* W_score = (const float*)d_in[4];
    const float* b_score = (const float*)d_in[5];
    float* out = (float*)d_out;
    float* sig = (float*)d_ws;   // 16 floats of scratch

    mqa_scores_kernel<<<1, 512, 0, stream>>>(cond, W_kv, W_score, b_score, sig);

    const int nvec4 = (BATCH * HW) / 4;  // 16384
    mqa_broadcast_kernel<<<nvec4 / 256, 256, 0, stream>>>(sig, (float4*)out);
    (void)in_sizes; (void)n_in; (void)out_size; (void)ws_size;
}